// GraphConvLayer_59665685676453
// MI455X (gfx1250) — compile-verified
//
#include <hip/hip_runtime.h>

typedef __attribute__((ext_vector_type(2))) float v2f;
typedef __attribute__((ext_vector_type(8))) float v8f;

#define NODES 50000
#define DIN   512
#define DOUT  256
#define EDGES 800000

// ---------------------------------------------------------------------------
// Kernel 1: supports = inputs @ weights  via V_WMMA_F32_16X16X4_F32 (fp32 in,
// fp32 accumulate -- no precision loss vs reference).
// Block = 128 threads = 4 waves. Block b computes rows [16b, 16b+16).
// Wave w computes the 4 column tiles [64w, 64w+64), reusing its A fragment
// across the 4 WMMAs of each k-step (A streamed from HBM exactly once).
//
// Fragment layouts per CDNA5 ISA:
//   A 16x4 f32 : lane L<16 holds A[M=L][K=k..k+1] in v[0..1], lanes 16..31
//                hold K=k+2..k+3  -> contiguous v2f load per lane.
//   B 4x16 f32 : VGPR0 = B[k+2*half][n+lane%16], VGPR1 = B[k+1+2*half][...].
//   C/D 16x16  : VGPR r = row (r + 8*half), col = lane%16.
// ---------------------------------------------------------------------------
__global__ __launch_bounds__(128) void gemm_wmma_f32(const float* __restrict__ A,
                                                     const float* __restrict__ B,
                                                     float* __restrict__ C) {
  const int lane  = threadIdx.x & 31;
  const int wave  = threadIdx.x >> 5;
  const int half  = lane >> 4;   // which 16-lane half of the wave
  const int l15   = lane & 15;
  const int mbase = blockIdx.x * 16;
  const int nbase = wave * 64;

  v8f c0 = {}, c1 = {}, c2 = {}, c3 = {};

  const float* arow = A + (size_t)(mbase + l15) * DIN + 2 * half;        // per-lane A row
  const float* bcol = B + (size_t)(2 * half) * DOUT + nbase + l15;       // per-lane B col

  for (int k = 0; k < DIN; k += 4) {
    v2f a = *(const v2f*)(arow + k);                 // A[M=l15][k+2h .. k+2h+1]

    const float* bk = bcol + (size_t)k * DOUT;
    v2f b0, b1, b2, b3;
    b0.x = bk[0];   b0.y = bk[DOUT + 0];             // B[k+2h][n], B[k+2h+1][n]
    b1.x = bk[16];  b1.y = bk[DOUT + 16];
    b2.x = bk[32];  b2.y = bk[DOUT + 32];
    b3.x = bk[48];  b3.y = bk[DOUT + 48];

    // 8 args: (neg_a, A, neg_b, B, c_mod, C, reuse_a, reuse_b)
    c0 = __builtin_amdgcn_wmma_f32_16x16x4_f32(false, a, false, b0, (short)0, c0, false, false);
    c1 = __builtin_amdgcn_wmma_f32_16x16x4_f32(false, a, false, b1, (short)0, c1, false, false);
    c2 = __builtin_amdgcn_wmma_f32_16x16x4_f32(false, a, false, b2, (short)0, c2, false, false);
    c3 = __builtin_amdgcn_wmma_f32_16x16x4_f32(false, a, false, b3, (short)0, c3, false, false);
  }

  // D layout: VGPR r -> row (mbase + r + 8*half), col (nbase + l15 + 16*t)
  float* crow = C + (size_t)(mbase + 8 * half) * DOUT + nbase + l15;
#pragma unroll
  for (int r = 0; r < 8; ++r) {
    crow[(size_t)r * DOUT + 0]  = c0[r];
    crow[(size_t)r * DOUT + 16] = c1[r];
    crow[(size_t)r * DOUT + 32] = c2[r];
    crow[(size_t)r * DOUT + 48] = c3[r];
  }
}

// ---------------------------------------------------------------------------
// Kernel 2: out[i] = bias[i % 256]  (initializes the accumulation target;
// bias broadcast hits WGP$/L2, store is fully coalesced).
// NODES*DOUT is divisible by 256, grid is exact -> no bounds check.
// ---------------------------------------------------------------------------
__global__ __launch_bounds__(256) void init_bias(float* __restrict__ out,
                                                 const float* __restrict__ bias) {
  const int i = blockIdx.x * 256 + threadIdx.x;
  out[i] = bias[i & (DOUT - 1)];
}

// ---------------------------------------------------------------------------
// Kernel 3: COO scatter. One wave per edge: each lane handles 8 floats as
// two float4 (b128) gathers from supports[col] and 8 global_atomic_add_f32
// into out[row]. Edge metadata loads are wave-uniform -> scalarizable.
// EDGES divisible by 8 (waves/block) -> exact grid.
// ---------------------------------------------------------------------------
__global__ __launch_bounds__(256) void edge_scatter(const float* __restrict__ sup,
                                                    const float* __restrict__ ev,
                                                    const int*   __restrict__ er,
                                                    const int*   __restrict__ ec,
                                                    float* __restrict__ out) {
  const int lane = threadIdx.x & 31;
  const int wave = threadIdx.x >> 5;
  const int e    = blockIdx.x * 8 + wave;
  if (e >= EDGES) return;

  const float  v = ev[e];
  const size_t c = (size_t)ec[e];
  const size_t r = (size_t)er[e];

  const float4* src = (const float4*)(sup + c * DOUT);  // 64 float4 per row
  float*        dst = out + r * DOUT;

#pragma unroll
  for (int it = 0; it < 2; ++it) {
    const int idx = lane + it * 32;        // float4 index within the row
    const float4 s = src[idx];
    const int b = idx * 4;
    atomicAdd(dst + b + 0, s.x * v);
    atomicAdd(dst + b + 1, s.y * v);
    atomicAdd(dst + b + 2, s.z * v);
    atomicAdd(dst + b + 3, s.w * v);
  }
}

// ---------------------------------------------------------------------------
// inputs (dict order): inputs[50000x512] f32, weights[512x256] f32,
// bias[256] f32, edge_vals[800000] f32, edge_row[800000] i32,
// edge_col[800000] i32.  out: [50000x256] f32.
// workspace: supports = 50000*256*4 = 51.2 MB.
// ---------------------------------------------------------------------------
extern "C" void kernel_launch(void* const* d_in, const int* in_sizes, int n_in,
                              void* d_out, int out_size, void* d_ws, size_t ws_size,
                              hipStream_t stream) {
  const float* inputs  = (const float*)d_in[0];
  const float* weights = (const float*)d_in[1];
  const float* bias    = (const float*)d_in[2];
  const float* evals   = (const float*)d_in[3];
  const int*   erow    = (const int*)d_in[4];
  const int*   ecol    = (const int*)d_in[5];
  float* out      = (float*)d_out;
  float* supports = (float*)d_ws;   // NODES*DOUT floats

  gemm_wmma_f32<<<NODES / 16, 128, 0, stream>>>(inputs, weights, supports);
  init_bias<<<(NODES * DOUT) / 256, 256, 0, stream>>>(out, bias);
  edge_scatter<<<EDGES / 8, 256, 0, stream>>>(supports, evals, erow, ecol, out);
}